// DiffusionFlowEmbedder_76081050681552
// MI455X (gfx1250) — compile-verified
//
#include <hip/hip_runtime.h>
#include <hip/hip_bf16.h>
#include <math.h>

#define N_PTS 2048
#define DIM 100
#define EMBD 2
#define SIGMA_G 0.5f
#define SIGMA_E 0.5f
#define FS_GRAPH 5.0f
#define EPSN 1e-12f

typedef __attribute__((ext_vector_type(2))) float v2f;
typedef __attribute__((ext_vector_type(8))) float v8f;

// ---------------------------------------------------------------------------
// f32 WMMA GEMM: C = A * B, 2048x2048x2048, row-major.
// Block = 128 threads (4 waves, arranged 2x2). Block tile = 64 rows x 128 cols.
// Each wave computes a 32x64 register tile: 2 M-tiles x 4 N-tiles = 8
// accumulators; per K=4 step, 2 A-fragments + 4 B-fragments feed 8 WMMAs
// (A reused across the 4 N-tiles, B reused across the 2 M-tiles).
// V_WMMA_F32_16X16X4_F32 layout (ISA 7.12.2):
//   A (16x4): lane<16 -> M=lane, K={k,k+1}; lane>=16 -> M=lane-16, K={k+2,k+3}
//   B (4x16): symmetric with N in place of M
//   C/D: 8 VGPRs; lanes 0-15 N=lane M=vgpr, lanes 16-31 N=lane-16 M=vgpr+8
// ---------------------------------------------------------------------------
__global__ __launch_bounds__(128) void gemm2048_wmma(const float* __restrict__ A,
                                                     const float* __restrict__ B,
                                                     float* __restrict__ C) {
  const int lane = threadIdx.x & 31;
  const int wave = threadIdx.x >> 5;      // 0..3
  const int half = lane >> 4;             // 0/1
  const int ln   = lane & 15;
  const int row0 = blockIdx.y * 64 + (wave >> 1) * 32;   // wave: 32 rows
  const int col0 = blockIdx.x * 128 + (wave & 1) * 64;   // wave: 64 cols

  const int m0 = row0 + ln;
  const int m1 = m0 + 16;

  v8f acc[2][4];
#pragma unroll
  for (int i = 0; i < 2; ++i)
#pragma unroll
    for (int j = 0; j < 4; ++j) acc[i][j] = (v8f){};

  const float* __restrict__ Arow0 = A + (size_t)m0 * N_PTS;
  const float* __restrict__ Arow1 = A + (size_t)m1 * N_PTS;

#pragma unroll 2
  for (int k = 0; k < N_PTS; k += 4) {
    const int kb = k + half * 2;
    v2f a0, a1;
    a0.x = Arow0[kb];
    a0.y = Arow0[kb + 1];
    a1.x = Arow1[kb];
    a1.y = Arow1[kb + 1];
    const float* __restrict__ Bk = B + (size_t)kb * N_PTS;
    v2f b[4];
#pragma unroll
    for (int j = 0; j < 4; ++j) {
      const int n = col0 + j * 16 + ln;
      b[j].x = Bk[n];
      b[j].y = Bk[N_PTS + n];
    }
#pragma unroll
    for (int j = 0; j < 4; ++j) {
      acc[0][j] = __builtin_amdgcn_wmma_f32_16x16x4_f32(false, a0, false, b[j],
                                                        (short)0, acc[0][j], false, false);
      acc[1][j] = __builtin_amdgcn_wmma_f32_16x16x4_f32(false, a1, false, b[j],
                                                        (short)0, acc[1][j], false, false);
    }
  }

#pragma unroll
  for (int i = 0; i < 2; ++i) {
    float* __restrict__ Crow = C + (size_t)(row0 + i * 16 + half * 8) * N_PTS;
#pragma unroll
    for (int r = 0; r < 8; ++r) {
#pragma unroll
      for (int j = 0; j < 4; ++j) {
        Crow[(size_t)r * N_PTS + col0 + j * 16 + ln] = acc[i][j][r];
      }
    }
  }
}

// ---------------------------------------------------------------------------
// Normalize flows per row; fsq[i] = ||fl_i||^2 (matches reference exactly).
// ---------------------------------------------------------------------------
__global__ void norm_flows(const float* __restrict__ flows,
                           float* __restrict__ fl, float* __restrict__ fsq) {
  int i = blockIdx.x * blockDim.x + threadIdx.x;
  if (i >= N_PTS) return;
  const float* src = flows + (size_t)i * DIM;
  float s = 0.f;
  for (int d = 0; d < DIM; ++d) s += src[d] * src[d];
  float inv = 1.f / fmaxf(sqrtf(s), EPSN);
  float q = 0.f;
  float* dst = fl + (size_t)i * DIM;
  for (int d = 0; d < DIM; ++d) {
    float v = src[d] * inv;
    dst[d] = v;
    q += v * v;
  }
  fsq[i] = q;
}

// ---------------------------------------------------------------------------
// Graph affinity: Pg[i][j] = exp(-(FS*|fsq_i - nd_ij.fl_i| + len_ij)/sigma)
// 16x16 output tile per 256-thread block; rows staged in LDS (~19 KB).
// ---------------------------------------------------------------------------
__global__ __launch_bounds__(256) void affinity_graph(const float* __restrict__ X,
                                                      const float* __restrict__ fl,
                                                      const float* __restrict__ fsq,
                                                      float* __restrict__ Pg) {
  __shared__ float sXi[16][DIM];
  __shared__ float sXj[16][DIM];
  __shared__ float sFl[16][DIM];
  __shared__ float sFq[16];
  const int i0 = blockIdx.y * 16;
  const int j0 = blockIdx.x * 16;
  for (int t = threadIdx.x; t < 16 * DIM; t += 256) {
    int r = t / DIM, c = t % DIM;
    sXi[r][c] = X[(size_t)(i0 + r) * DIM + c];
    sXj[r][c] = X[(size_t)(j0 + r) * DIM + c];
    sFl[r][c] = fl[(size_t)(i0 + r) * DIM + c];
  }
  if (threadIdx.x < 16) sFq[threadIdx.x] = fsq[i0 + threadIdx.x];
  __syncthreads();

  const int ti = threadIdx.x >> 4;
  const int tj = threadIdx.x & 15;
  float len2 = 0.f, dotr = 0.f;
#pragma unroll 4
  for (int d = 0; d < DIM; ++d) {
    float diff = sXj[tj][d] - sXi[ti][d];
    len2 += diff * diff;
    dotr += diff * sFl[ti][d];
  }
  float len  = sqrtf(len2);
  float dot  = dotr / fmaxf(len, EPSN);
  float dist = FS_GRAPH * fabsf(sFq[ti] - dot) + len;
  Pg[(size_t)(i0 + ti) * N_PTS + (j0 + tj)] = expf(-dist / SIGMA_G);
}

// ---------------------------------------------------------------------------
// Encoder MLP + flow artist, one thread per row. Weights [fi,fo] row-major.
// ---------------------------------------------------------------------------
__global__ void encoder_flow(const float* __restrict__ X,
                             const float* eW0, const float* eb0,
                             const float* eW1, const float* eb1,
                             const float* eW2, const float* eb2,
                             const float* fW0, const float* fb0,
                             const float* fW1, const float* fb1,
                             const float* fW2, const float* fb2,
                             const float* fW3, const float* fb3,
                             float* __restrict__ emb, float* __restrict__ flow_e) {
  int i = blockIdx.x * blockDim.x + threadIdx.x;
  if (i >= N_PTS) return;
  const float* x = X + (size_t)i * DIM;
  float h1[100];
  for (int o = 0; o < 100; ++o) {
    float s = eb0[o];
    for (int d = 0; d < DIM; ++d) s += x[d] * eW0[d * 100 + o];
    h1[o] = fmaxf(s, 0.f);
  }
  float h2[10];
  for (int o = 0; o < 10; ++o) {
    float s = eb1[o];
    for (int d = 0; d < 100; ++d) s += h1[d] * eW1[d * 10 + o];
    h2[o] = fmaxf(s, 0.f);
  }
  float e[EMBD];
  for (int o = 0; o < EMBD; ++o) {
    float s = eb2[o];
    for (int d = 0; d < 10; ++d) s += h2[d] * eW2[d * EMBD + o];
    e[o] = s;
    emb[(size_t)i * EMBD + o] = s;
  }
  float f1[10];
  for (int o = 0; o < 10; ++o) {
    float s = fb0[o];
    for (int d = 0; d < EMBD; ++d) s += e[d] * fW0[d * 10 + o];
    f1[o] = tanhf(s);
  }
  float f2[20];
  for (int o = 0; o < 20; ++o) {
    float s = fb1[o];
    for (int d = 0; d < 10; ++d) s += f1[d] * fW1[d * 20 + o];
    f2[o] = tanhf(s);
  }
  float f3[10];
  for (int o = 0; o < 10; ++o) {
    float s = fb2[o];
    for (int d = 0; d < 20; ++d) s += f2[d] * fW2[d * 10 + o];
    f3[o] = tanhf(s);
  }
  for (int o = 0; o < EMBD; ++o) {
    float s = fb3[o];
    for (int d = 0; d < 10; ++d) s += f3[d] * fW3[d * EMBD + o];
    flow_e[(size_t)i * EMBD + o] = s;
  }
}

// ---------------------------------------------------------------------------
// Embedding affinity + row normalization: one block per row.
// ---------------------------------------------------------------------------
__global__ __launch_bounds__(256) void pe_rows(const float* __restrict__ emb,
                                               const float* __restrict__ flow_e,
                                               const float* __restrict__ fs_ptr,
                                               float* __restrict__ Pe) {
  __shared__ float sred[256];
  const int i = blockIdx.x;
  const float fs = fs_ptr[0];
  const float ei0 = emb[i * 2], ei1 = emb[i * 2 + 1];
  float f0 = flow_e[i * 2], f1 = flow_e[i * 2 + 1];
  float inv = 1.f / fmaxf(sqrtf(f0 * f0 + f1 * f1), EPSN);
  f0 *= inv; f1 *= inv;
  const float fsq = f0 * f0 + f1 * f1;

  float lsum = 0.f;
  for (int j = threadIdx.x; j < N_PTS; j += 256) {
    float d0 = emb[j * 2] - ei0;
    float d1 = emb[j * 2 + 1] - ei1;
    float len = sqrtf(d0 * d0 + d1 * d1);
    float il  = 1.f / fmaxf(len, EPSN);
    float dot = (d0 * f0 + d1 * f1) * il;
    float a = expf(-(fs * fabsf(fsq - dot) + len) / SIGMA_E);
    Pe[(size_t)i * N_PTS + j] = a;
    lsum += a;
  }
  sred[threadIdx.x] = lsum;
  __syncthreads();
  for (int s = 128; s > 0; s >>= 1) {
    if (threadIdx.x < s) sred[threadIdx.x] += sred[threadIdx.x + s];
    __syncthreads();
  }
  const float rinv = 1.f / sred[0];
  for (int j = threadIdx.x; j < N_PTS; j += 256)
    Pe[(size_t)i * N_PTS + j] *= rinv;
}

// ---------------------------------------------------------------------------
// Decoder + reconstruction SSE (accum[0]).
// ---------------------------------------------------------------------------
__global__ void decoder_recon(const float* __restrict__ X, const float* __restrict__ emb,
                              const float* dW0, const float* db0,
                              const float* dW1, const float* db1,
                              const float* dW2, const float* db2,
                              float* __restrict__ accum) {
  int i = blockIdx.x * blockDim.x + threadIdx.x;
  if (i >= N_PTS) return;
  const float e0 = emb[i * 2], e1 = emb[i * 2 + 1];
  float g1[10];
  for (int o = 0; o < 10; ++o)
    g1[o] = fmaxf(db0[o] + e0 * dW0[o] + e1 * dW0[10 + o], 0.f);
  float g2[100];
  for (int o = 0; o < 100; ++o) {
    float s = db1[o];
    for (int d = 0; d < 10; ++d) s += g1[d] * dW1[d * 100 + o];
    g2[o] = fmaxf(s, 0.f);
  }
  float sse = 0.f;
  const float* x = X + (size_t)i * DIM;
  for (int d = 0; d < DIM; ++d) {
    float s = db2[d];
    for (int o = 0; o < 100; ++o) s += g2[o] * dW2[o * 100 + d];
    float diff = s - x[d];
    sse += diff * diff;
  }
  atomicAdd(&accum[0], sse);
}

// ---------------------------------------------------------------------------
// KLD = sum(xlogy(t,t) - t*log(p)) into accum[1].
// ---------------------------------------------------------------------------
__global__ __launch_bounds__(256) void kld_reduce(const float* __restrict__ Pg4,
                                                  const float* __restrict__ Pe4,
                                                  float* __restrict__ accum) {
  __shared__ float sred[256];
  float lsum = 0.f;
  const size_t total = (size_t)N_PTS * N_PTS;
  for (size_t idx = (size_t)blockIdx.x * 256u + threadIdx.x; idx < total;
       idx += (size_t)gridDim.x * 256u) {
    float t = Pg4[idx];
    float p = Pe4[idx];
    float xl = (t > 0.f) ? t * logf(t) : 0.f;
    lsum += xl - t * logf(p);
  }
  sred[threadIdx.x] = lsum;
  __syncthreads();
  for (int s = 128; s > 0; s >>= 1) {
    if (threadIdx.x < s) sred[threadIdx.x] += sred[threadIdx.x + s];
    __syncthreads();
  }
  if (threadIdx.x == 0) atomicAdd(&accum[1], sred[0]);
}

__global__ void init_accum(float* accum) {
  accum[0] = 0.f;
  accum[1] = 0.f;
}

__global__ void finalize(const float* __restrict__ accum, float* __restrict__ out) {
  out[0] = accum[1] / (float)N_PTS + accum[0] / (float)(N_PTS * DIM);
}

// ---------------------------------------------------------------------------
extern "C" void kernel_launch(void* const* d_in, const int* in_sizes, int n_in,
                              void* d_out, int out_size, void* d_ws, size_t ws_size,
                              hipStream_t stream) {
  const float* X     = (const float*)d_in[0];
  const float* flows = (const float*)d_in[1];
  const float* eW0 = (const float*)d_in[2];  const float* eb0 = (const float*)d_in[3];
  const float* eW1 = (const float*)d_in[4];  const float* eb1 = (const float*)d_in[5];
  const float* eW2 = (const float*)d_in[6];  const float* eb2 = (const float*)d_in[7];
  const float* dW0 = (const float*)d_in[8];  const float* db0 = (const float*)d_in[9];
  const float* dW1 = (const float*)d_in[10]; const float* db1 = (const float*)d_in[11];
  const float* dW2 = (const float*)d_in[12]; const float* db2 = (const float*)d_in[13];
  const float* fW0 = (const float*)d_in[14]; const float* fb0 = (const float*)d_in[15];
  const float* fW1 = (const float*)d_in[16]; const float* fb1 = (const float*)d_in[17];
  const float* fW2 = (const float*)d_in[18]; const float* fb2 = (const float*)d_in[19];
  const float* fW3 = (const float*)d_in[20]; const float* fb3 = (const float*)d_in[21];
  const float* fs  = (const float*)d_in[22];

  char* ws = (char*)d_ws;
  const size_t MATB = (size_t)N_PTS * N_PTS * sizeof(float);   // 16 MB
  float* B0 = (float*)(ws);                 // Pg   -> later Pe2
  float* B1 = (float*)(ws + MATB);          // Pg2  -> later Pe, Pe4
  float* B2 = (float*)(ws + 2 * MATB);      // Pg4 (kept)
  float* fl     = (float*)(ws + 3 * MATB);
  float* fsq    = fl + (size_t)N_PTS * DIM;
  float* emb    = fsq + N_PTS;
  float* flow_e = emb + (size_t)N_PTS * EMBD;
  float* accum  = flow_e + (size_t)N_PTS * EMBD;

  init_accum<<<1, 1, 0, stream>>>(accum);

  norm_flows<<<(N_PTS + 127) / 128, 128, 0, stream>>>(flows, fl, fsq);

  dim3 gA(N_PTS / 16, N_PTS / 16);
  affinity_graph<<<gA, 256, 0, stream>>>(X, fl, fsq, B0);           // Pg -> B0

  dim3 gG(N_PTS / 128, N_PTS / 64);
  gemm2048_wmma<<<gG, 128, 0, stream>>>(B0, B0, B1);                // Pg^2 -> B1
  gemm2048_wmma<<<gG, 128, 0, stream>>>(B1, B1, B2);                // Pg^4 -> B2

  encoder_flow<<<(N_PTS + 127) / 128, 128, 0, stream>>>(
      X, eW0, eb0, eW1, eb1, eW2, eb2,
      fW0, fb0, fW1, fb1, fW2, fb2, fW3, fb3, emb, flow_e);

  pe_rows<<<N_PTS, 256, 0, stream>>>(emb, flow_e, fs, B1);          // Pe -> B1
  gemm2048_wmma<<<gG, 128, 0, stream>>>(B1, B1, B0);                // Pe^2 -> B0
  gemm2048_wmma<<<gG, 128, 0, stream>>>(B0, B0, B1);                // Pe^4 -> B1

  decoder_recon<<<(N_PTS + 127) / 128, 128, 0, stream>>>(
      X, emb, dW0, db0, dW1, db1, dW2, db2, accum);

  kld_reduce<<<1024, 256, 0, stream>>>(B2, B1, accum);
  finalize<<<1, 1, 0, stream>>>(accum, (float*)d_out);
}